// _ProposalLayer_70755291234457
// MI455X (gfx1250) — compile-verified
//
#include <hip/hip_runtime.h>
#include <hip/hip_bf16.h>

#define A_NUM   9
#define H_DIM   64
#define W_DIM   64
#define N_PROP  (H_DIM * W_DIM * A_NUM)   // 36864
#define B_NUM   2
#define TOP_N   2000
#define NMS_T   0.7f
#define NEGV    (-1.0e30f)
#define FSTRIDE 16.0f
#define NMS_THREADS 1024
#define IPT (N_PROP / NMS_THREADS)        // 36 items per thread
#define FULL_MASK ((1ull << IPT) - 1ull)

// canonical Faster-RCNN anchors: base 16, ratios {0.5,1,2}, scales {8,16,32}
__constant__ float c_anc[A_NUM][4] = {
    { -84.f,  -40.f,  99.f,  55.f},
    {-176.f,  -88.f, 191.f, 103.f},
    {-360.f, -184.f, 375.f, 199.f},
    { -56.f,  -56.f,  71.f,  71.f},
    {-120.f, -120.f, 135.f, 135.f},
    {-248.f, -248.f, 263.f, 263.f},
    { -36.f,  -84.f,  51.f,  99.f},
    { -80.f, -176.f,  95.f, 191.f},
    {-168.f, -360.f, 183.f, 375.f}
};

// ---------------- kernel 1: anchors + bbox_transform_inv + clip ----------------
__global__ void proposal_gen(const float* __restrict__ score_map,   // (B, 2A, H, W)
                             const float* __restrict__ bbox_delta,  // (B, 4A, H, W)
                             const float* __restrict__ img_info,    // (B, 3)
                             float4* __restrict__ boxes,            // (B*N)
                             float* __restrict__ scores)            // (B*N)
{
    int idx = blockIdx.x * blockDim.x + threadIdx.x;
    if (idx >= B_NUM * N_PROP) return;
    int b  = idx / N_PROP;
    int p  = idx - b * N_PROP;          // p = (h*W + w)*A + a
    int a  = p % A_NUM;
    int hw = p / A_NUM;
    int w  = hw % W_DIM;
    int h  = hw / W_DIM;

    const int plane = H_DIM * W_DIM;
    // fg score: channel A + a
    float s = score_map[((b * 2 * A_NUM + A_NUM + a) * H_DIM + h) * W_DIM + w];
    // deltas: channels 4a..4a+3
    int dbase = ((b * 4 * A_NUM + 4 * a) * H_DIM + h) * W_DIM + w;
    float dx = bbox_delta[dbase];
    float dy = bbox_delta[dbase + plane];
    float dw = bbox_delta[dbase + 2 * plane];
    float dh = bbox_delta[dbase + 3 * plane];

    float cx = ((float)w + 0.5f) * FSTRIDE;
    float cy = ((float)h + 0.5f) * FSTRIDE;
    float ax1 = cx + c_anc[a][0];
    float ay1 = cy + c_anc[a][1];
    float ax2 = cx + c_anc[a][2];
    float ay2 = cy + c_anc[a][3];

    float aw  = ax2 - ax1 + 1.0f;
    float ah  = ay2 - ay1 + 1.0f;
    float acx = ax1 + 0.5f * aw;
    float acy = ay1 + 0.5f * ah;

    float pcx = dx * aw + acx;
    float pcy = dy * ah + acy;
    float pw  = __expf(dw) * aw;
    float ph  = __expf(dh) * ah;

    float x1 = pcx - 0.5f * pw;
    float y1 = pcy - 0.5f * ph;
    float x2 = pcx + 0.5f * pw;
    float y2 = pcy + 0.5f * ph;

    float imh = img_info[b * 3 + 0] - 1.0f;
    float imw = img_info[b * 3 + 1] - 1.0f;
    x1 = fminf(fmaxf(x1, 0.0f), imw);
    x2 = fminf(fmaxf(x2, 0.0f), imw);
    y1 = fminf(fmaxf(y1, 0.0f), imh);
    y2 = fminf(fmaxf(y2, 0.0f), imh);

    boxes[idx]  = make_float4(x1, y1, x2, y2);
    scores[idx] = s;
}

// ---------------- kernel 2: greedy NMS, one workgroup per image ----------------
__global__ __launch_bounds__(NMS_THREADS, 1)
void nms_kernel(const float4* __restrict__ boxes_all,
                const float* __restrict__ scores_all,
                float* __restrict__ out)
{
    const int b = blockIdx.x;
    const float4* __restrict__ boxes  = boxes_all  + b * N_PROP;
    const float*  __restrict__ scores = scores_all + b * N_PROP;
    float* __restrict__ outB = out + b * TOP_N * 5;

    __shared__ float s_sc[N_PROP];   // 144 KB score field, LDS-resident (read-only after init)
    __shared__ float s_rv[32];
    __shared__ int   s_ri[32];
    __shared__ int   s_bi;
    __shared__ float s_bv;

    const int t    = threadIdx.x;
    const int wave = t >> 5;
    const int lane = t & 31;

    // ---- stage scores into LDS with CDNA5 async global->LDS loads ----
    {
        unsigned lds_base = (unsigned)(size_t)(&s_sc[0]);
        #pragma unroll
        for (int j = 0; j < IPT; ++j) {
            int p = t + j * NMS_THREADS;
            unsigned loff = lds_base + (unsigned)p * 4u;
            const float* g = scores + p;
            asm volatile("global_load_async_to_lds_b32 %0, %1, off"
                         :: "v"(loff), "v"(g) : "memory");
        }
        asm volatile("s_wait_asynccnt 0" ::: "memory");
    }
    __syncthreads();

    // per-thread liveness mask: bit j <=> item p = t + j*1024 still unsuppressed
    unsigned long long alive = FULL_MASK;

    // ---- initial per-thread argmax over owned slots ----
    float lv = -3.402823466e38f;
    int   li = 0;
    #pragma unroll
    for (int j = 0; j < IPT; ++j) {
        int p = t + j * NMS_THREADS;
        float s = s_sc[p];
        if (s > lv) { lv = s; li = p; }
    }

    for (int it = 0; it < TOP_N; ++it) {
        // wave32 argmax reduce (prefer lower index on ties, like jnp.argmax)
        float v = lv; int i0 = li;
        #pragma unroll
        for (int off = 16; off > 0; off >>= 1) {
            float ov = __shfl_xor(v, off, 32);
            int   oi = __shfl_xor(i0, off, 32);
            if (ov > v || (ov == v && oi < i0)) { v = ov; i0 = oi; }
        }
        if (lane == 0) { s_rv[wave] = v; s_ri[wave] = i0; }
        __syncthreads();
        if (wave == 0) {
            float v2 = s_rv[lane]; int i2 = s_ri[lane];
            #pragma unroll
            for (int off = 16; off > 0; off >>= 1) {
                float ov = __shfl_xor(v2, off, 32);
                int   oi = __shfl_xor(i2, off, 32);
                if (ov > v2 || (ov == v2 && oi < i2)) { v2 = ov; i2 = oi; }
            }
            if (lane == 0) { s_bi = i2; s_bv = v2; }
        }
        __syncthreads();
        const int   isel = s_bi;
        const float mv   = s_bv;

        if (!(mv > 0.5f * NEGV)) {
            // everything suppressed: remaining rows are zeros (matches reference)
            for (int k = it * 5 + t; k < TOP_N * 5; k += NMS_THREADS)
                outB[k] = 0.0f;
            return;
        }

        float4 bb = boxes[isel];
        if (t == 0) {
            outB[it * 5 + 0] = scores[isel];   // original (unsuppressed) score
            outB[it * 5 + 1] = bb.x;
            outB[it * 5 + 2] = bb.y;
            outB[it * 5 + 3] = bb.z;
            outB[it * 5 + 4] = bb.w;
        }
        const float barea = (bb.z - bb.x + 1.0f) * (bb.w - bb.y + 1.0f);

        // suppression fused with local-argmax recompute: visit only live items.
        // Bits iterate LSB->MSB == increasing p, so strict '>' keeps lowest index.
        lv = -3.402823466e38f;
        li = 0;
        unsigned long long m = alive;
        while (m) {
            int j = (int)__builtin_ctzll(m);
            m &= m - 1;
            int p = t + j * NMS_THREADS;
            float4 pb = boxes[p];              // L2-resident (576 KB/image)
            float iw = fminf(pb.z, bb.z) - fmaxf(pb.x, bb.x) + 1.0f;
            float ih = fminf(pb.w, bb.w) - fmaxf(pb.y, bb.y) + 1.0f;
            iw = fmaxf(iw, 0.0f);
            ih = fmaxf(ih, 0.0f);
            float inter = iw * ih;
            float parea = (pb.z - pb.x + 1.0f) * (pb.w - pb.y + 1.0f);
            // iou > T  <=>  inter > T*(a1+a2-inter), division-free
            if (inter > NMS_T * (parea + barea - inter)) {
                alive &= ~(1ull << j);         // dead: never touched again
            } else {
                float s = s_sc[p];
                if (s > lv) { lv = s; li = p; }
            }
        }
        // no barrier needed here: s_sc is read-only, masks are thread-private,
        // and s_rv/s_bi hazards are covered by the two barriers above.
    }
}

extern "C" void kernel_launch(void* const* d_in, const int* in_sizes, int n_in,
                              void* d_out, int out_size, void* d_ws, size_t ws_size,
                              hipStream_t stream) {
    const float* score_map  = (const float*)d_in[0];   // (2, 18, 64, 64) f32
    const float* bbox_delta = (const float*)d_in[1];   // (2, 36, 64, 64) f32
    const float* img_info   = (const float*)d_in[2];   // (2, 3) f32
    float* out = (float*)d_out;                        // (2, 2000, 5) f32

    float4* boxes  = (float4*)d_ws;                                           // B*N float4
    float*  scores = (float*)((char*)d_ws + sizeof(float4) * B_NUM * N_PROP); // B*N f32

    int total = B_NUM * N_PROP;
    proposal_gen<<<(total + 255) / 256, 256, 0, stream>>>(
        score_map, bbox_delta, img_info, boxes, scores);
    nms_kernel<<<B_NUM, NMS_THREADS, 0, stream>>>(boxes, scores, out);
}